// ENT_69853348102921
// MI455X (gfx1250) — compile-verified
//
#include <hip/hip_runtime.h>
#include <stdint.h>

// Problem geometry (fixed by the reference).
#define NSLICES     8
#define SLICE_VALS  (128 * 1024)      // 131072 downsampled values per slice
#define PLANE_ELEMS (256 * 256)
#define VALS_TOTAL  (NSLICES * SLICE_VALS)   // 1,048,576
#define BINS        30

// ---- monotonic float <-> uint mapping for atomic min/max --------------------
__device__ __forceinline__ uint32_t fmap(float f) {
    uint32_t u = __float_as_uint(f);
    return (u & 0x80000000u) ? ~u : (u | 0x80000000u);
}
__device__ __forceinline__ float funmap(uint32_t m) {
    uint32_t u = (m & 0x80000000u) ? (m ^ 0x80000000u) : ~m;
    return __uint_as_float(u);
}

// slice s -> first plane index (contiguous 128 planes per slice)
__device__ __forceinline__ int slice_base_plane(int s) {
    return ((s >> 1) << 10) + ((s & 1) << 9);   // b*1024 + half*512
}

// ----------------------------------------------------------------------------
// Core: one 256-thread block computes 256 downsampled values (= quarter of one
// 32x32 plane). The 16 needed source rows (rows 8y+3 and 8y+4 for 8 y's) are
// staged into LDS with CDNA5 async tensor-path loads
// (global_load_async_to_lds_b128, tracked by ASYNCcnt), then each thread does
// the exact 4-tap bilinear average from LDS.
// ----------------------------------------------------------------------------
__device__ __forceinline__ float downsample_val(const float* __restrict__ in,
                                                float* smem, int blockId, int tid) {
    const int gid0 = blockId << 8;                  // first value index of block
    const int s    = gid0 >> 17;                    // slice (131072 = 2^17)
    const int k0   = gid0 & (SLICE_VALS - 1);
    const int j    = k0 >> 10;                      // plane within slice
    const int q    = (k0 >> 8) & 3;                 // quarter of the 32x32 plane
    const int plane = slice_base_plane(s) + j;
    const float* pbase = in + (size_t)plane * PLANE_ELEMS;

    const uint32_t lds_base = (uint32_t)(uintptr_t)smem;  // low 32 bits = LDS offset
    #pragma unroll
    for (int i = 0; i < 4; ++i) {
        const int id = (i << 8) + tid;              // 0..1023 staged 16B chunks
        const int r  = id >> 6;                     // staged row 0..15
        const int c  = id & 63;                     // 16B chunk within row
        const int y  = (q << 3) + (r >> 1);         // output row 0..31
        const int row = (y << 3) + 3 + (r & 1);     // source row 8y+3 / 8y+4
        const uint64_t gp = (uint64_t)(uintptr_t)(pbase + row * 256 + (c << 2));
        const uint32_t lo = lds_base + (uint32_t)(id << 4);
        asm volatile("global_load_async_to_lds_b128 %0, %1, off"
                     :: "v"(lo), "v"(gp) : "memory");
    }
    asm volatile("s_wait_asynccnt 0" ::: "memory");  // this wave's DMA done
    __syncthreads();                                  // all waves' LDS visible

    const int yl  = tid >> 5;            // 0..7 (local output row)
    const int x   = tid & 31;            // output col
    const int rb  = (yl << 1) << 8;      // staged row pair base (floats)
    const int col = (x << 3) + 3;        // 8x+3
    const float v00 = smem[rb + col];
    const float v01 = smem[rb + col + 1];
    const float v10 = smem[rb + 256 + col];
    const float v11 = smem[rb + 256 + col + 1];
    // exact pairing: W-dim pair sums, then H-dim sum, exact *0.25 scale
    return ((v00 + v01) + (v10 + v11)) * 0.25f;
}

// ---- kernel 0: init per-slice min/max + histogram counters ------------------
__global__ void ent_init(uint32_t* __restrict__ mm, uint32_t* __restrict__ counts) {
    const int tid = threadIdx.x;
    if (tid < NSLICES) { mm[2 * tid] = 0xFFFFFFFFu; mm[2 * tid + 1] = 0u; }
    if (tid < NSLICES * BINS) counts[tid] = 0u;
}

// ---- kernel 1: downsample + per-slice min/max (optionally store values) -----
template <bool STORE>
__global__ __launch_bounds__(256)
void ent_downsample_minmax(const float* __restrict__ in,
                           float* __restrict__ vals,
                           uint32_t* __restrict__ mm) {
    __shared__ __align__(16) float smem[16 * 256];   // 16 staged rows, 16 KB
    __shared__ uint32_t s_min, s_max;
    const int tid = threadIdx.x;
    if (tid == 0) { s_min = 0xFFFFFFFFu; s_max = 0u; }

    const float v = downsample_val(in, smem, blockIdx.x, tid); // barrier inside

    if constexpr (STORE) vals[(blockIdx.x << 8) + tid] = v;
    const uint32_t m = fmap(v);
    atomicMin(&s_min, m);
    atomicMax(&s_max, m);
    __syncthreads();
    if (tid == 0) {
        const int s = (blockIdx.x << 8) >> 17;
        atomicMin(&mm[2 * s], s_min);
        atomicMax(&mm[2 * s + 1], s_max);
    }
}

// ---- kernel 2: histogram with LDS privatization -----------------------------
template <bool HAS_VALS>
__global__ __launch_bounds__(256)
void ent_histogram(const float* __restrict__ in,
                   const float* __restrict__ vals,
                   const uint32_t* __restrict__ mm,
                   uint32_t* __restrict__ counts) {
    __shared__ uint32_t lh[BINS];
    const int tid = threadIdx.x;
    if (tid < BINS) lh[tid] = 0u;

    const int gid = (blockIdx.x << 8) + tid;
    const int s   = gid >> 17;

    float v;
    if constexpr (HAS_VALS) {
        v = vals[gid];
        __syncthreads();                       // cover lh init
    } else {
        __shared__ __align__(16) float smem[16 * 256];
        v = downsample_val(in, smem, blockIdx.x, tid);  // barrier inside
    }

    const float lo = funmap(mm[2 * s]);
    const float hi = funmap(mm[2 * s + 1]);
    // match jnp: clip(floor((v-lo)/(hi-lo)*30), 0, 29) in fp32
    float f = floorf((v - lo) / (hi - lo) * 30.0f);
    f = fminf(fmaxf(f, 0.0f), 29.0f);
    atomicAdd(&lh[(int)f], 1u);
    __syncthreads();
    if (tid < BINS) atomicAdd(&counts[s * BINS + tid], lh[tid]);
}

// ---- kernel 3: entropy per slice + final mean -------------------------------
__global__ void ent_finalize(const uint32_t* __restrict__ counts,
                             float* __restrict__ out) {
    __shared__ float ents[NSLICES];
    const int tid = threadIdx.x;
    if (tid < NSLICES) {
        float c[BINS];
        float mx = 0.0f;
        for (int b = 0; b < BINS; ++b) {
            c[b] = (float)counts[tid * BINS + b];
            mx = fmaxf(mx, c[b]);
        }
        const float denom = mx + 1e-4f;
        bool anyzero = false;
        float acc = 0.0f;
        for (int b = 0; b < BINS; ++b) {
            if (c[b] == 0.0f) anyzero = true;            // p==0 -> NaN -> MAX_ENT
            else { const float p = c[b] / denom; acc += p * log2f(p); }
        }
        ents[tid] = anyzero ? 1.0f : -acc;
    }
    __syncthreads();
    if (tid == 0) {
        float sum = 0.0f;
        for (int s = 0; s < NSLICES; ++s) sum += ents[s];
        out[0] = (sum / 8.0f) / 10.0f;                    // mean(ents)/10
    }
}

// ---- host launcher ----------------------------------------------------------
extern "C" void kernel_launch(void* const* d_in, const int* in_sizes, int n_in,
                              void* d_out, int out_size, void* d_ws, size_t ws_size,
                              hipStream_t stream) {
    const float* in = (const float*)d_in[0];
    uint8_t* ws = (uint8_t*)d_ws;
    uint32_t* mm     = (uint32_t*)ws;           // 16 u32: per-slice {min,max}
    uint32_t* counts = (uint32_t*)(ws + 256);   // 8*30 u32
    float*    vals   = (float*)(ws + 4096);     // 4 MB of downsampled values

    const bool has_vals = ws_size >= (size_t)4096 + (size_t)VALS_TOTAL * 4;
    const int blocks = VALS_TOTAL / 256;        // 4096

    ent_init<<<1, 256, 0, stream>>>(mm, counts);
    if (has_vals) {
        ent_downsample_minmax<true ><<<blocks, 256, 0, stream>>>(in, vals, mm);
        ent_histogram       <true ><<<blocks, 256, 0, stream>>>(in, vals, mm, counts);
    } else {
        ent_downsample_minmax<false><<<blocks, 256, 0, stream>>>(in, nullptr, mm);
        ent_histogram       <false><<<blocks, 256, 0, stream>>>(in, nullptr, mm, counts);
    }
    ent_finalize<<<1, 32, 0, stream>>>(counts, (float*)d_out);
}